// DynamicsBasic_77300821393444
// MI455X (gfx1250) — compile-verified
//
#include <hip/hip_runtime.h>

typedef float v2f __attribute__((ext_vector_type(2)));
typedef float v8f __attribute__((ext_vector_type(8)));

__device__ __forceinline__ float read_lane_f32(float v, int lane) {
    int i = __builtin_amdgcn_readlane(__builtin_bit_cast(int, v), lane);
    return __builtin_bit_cast(float, i);
}

// y = (W @ x_i) . (x_f - x_i), streamed via V_WMMA_F32_16X16X4_F32.
// One wave handles a 16-row stripe of W over a contiguous column chunk.
// A-tile = 16x4 block of W (b64/lane), B = x[c0..c0+3] broadcast across all
// 16 N-columns (b64/lane, halves duplicate -> coalesced broadcast), C
// accumulates per-row dot products. 16 redundant columns are free: BW-bound.
__global__ __launch_bounds__(256)
void bilinear_matvec_wmma_kernel(const float* __restrict__ x_i,
                                 const float* __restrict__ x_f,
                                 const float* __restrict__ W,
                                 float* __restrict__ ws,
                                 int N, int chunks) {
    const int lane = threadIdx.x & 31;
    const int half = lane >> 4;      // K-half: 0 -> K=0,1 ; 1 -> K=2,3
    const int m    = lane & 15;      // A-matrix row within stripe

    const int wave   = (blockIdx.x * (blockDim.x >> 5)) + (threadIdx.x >> 5);
    const int stripe = wave / chunks;
    const int chunk  = wave % chunks;

    const int r0         = stripe * 16;
    const int chunk_cols = N / chunks;          // 1024 for N=16384, chunks=16
    const int c0         = chunk * chunk_cols;

    const float* wp = W   + (size_t)(r0 + m) * (size_t)N + c0 + 2 * half;
    const float* xp = x_i + c0 + 2 * half;

    v8f acc0 = {};
    v8f acc1 = {};

    // Each inner iteration: 2x (b64 W-load + b64 x-load + wmma), K advances 8.
    const int SEG  = 16;                        // 128 floats/row per segment
    const int nseg = chunk_cols / (8 * SEG);

    for (int s = 0; s < nseg; ++s) {
        __builtin_prefetch(wp + SEG * 8, 0, 0); // global_prefetch_b8, next seg
#pragma unroll
        for (int it = 0; it < SEG; ++it) {
            v2f a0 = *(const v2f*)(wp);
            v2f b0 = *(const v2f*)(xp);
            v2f a1 = *(const v2f*)(wp + 4);
            v2f b1 = *(const v2f*)(xp + 4);
            acc0 = __builtin_amdgcn_wmma_f32_16x16x4_f32(
                false, a0, false, b0, (short)0, acc0, false, false);
            acc1 = __builtin_amdgcn_wmma_f32_16x16x4_f32(
                false, a1, false, b1, (short)0, acc1, false, false);
            wp += 8;
            xp += 8;
        }
    }

    v8f acc = acc0 + acc1;

    // C/D layout: VGPR v holds row r0+v (lanes 0-15) / r0+8+v (lanes 16-31);
    // every column identical, so each lane in a half holds the full 8 dots.
    const int rbase = r0 + half * 8;
    float partial = 0.0f;
#pragma unroll
    for (int v = 0; v < 8; ++v) {
        const int r = rbase + v;
        partial += acc[v] * (x_f[r] - x_i[r]);
    }

    // lanes within a half are duplicates: take one rep from each half.
    const float total = read_lane_f32(partial, 0) + read_lane_f32(partial, 16);
    if (lane == 0) ws[wave] = total;
}

// Deterministic fixed-order reduction of the per-wave partials.
__global__ __launch_bounds__(256)
void bilinear_reduce_kernel(const float* __restrict__ ws,
                            float* __restrict__ out, int n) {
    __shared__ float s[256];
    float acc = 0.0f;
    for (int i = threadIdx.x; i < n; i += 256) acc += ws[i];
    s[threadIdx.x] = acc;
    __syncthreads();
    for (int off = 128; off > 0; off >>= 1) {
        if ((int)threadIdx.x < off) s[threadIdx.x] += s[threadIdx.x + off];
        __syncthreads();
    }
    if (threadIdx.x == 0) out[0] = s[0];
}

extern "C" void kernel_launch(void* const* d_in, const int* in_sizes, int n_in,
                              void* d_out, int out_size, void* d_ws, size_t ws_size,
                              hipStream_t stream) {
    const float* x_i = (const float*)d_in[0];
    const float* x_f = (const float*)d_in[1];
    const float* W   = (const float*)d_in[2];
    // d_in[3] is t: analytically cancels (d/dt interp == x_f - x_i).

    const int N = in_sizes[0];          // 16384
    const int CHUNKS = 16;              // column chunks per 16-row stripe
    const int stripes = N / 16;
    const int total_waves = stripes * CHUNKS;   // 16384 waves
    const int threads = 256;                    // 8 waves per block
    const int blocks = (total_waves * 32) / threads;

    float* ws = (float*)d_ws;

    bilinear_matvec_wmma_kernel<<<blocks, threads, 0, stream>>>(
        x_i, x_f, W, ws, N, CHUNKS);
    bilinear_reduce_kernel<<<1, 256, 0, stream>>>(ws, (float*)d_out, total_waves);
}